// Tacotron2Loss_23424751633082
// MI455X (gfx1250) — compile-verified
//
#include <hip/hip_runtime.h>
#include <math.h>

// ---------------------------------------------------------------------------
// Tacotron2 loss on MI455X (gfx1250).
// Bandwidth-bound streaming reduction (~164 MB @ 23.3 TB/s ~= 7 us).
// Deterministic two-stage reduction: per-block partials in d_ws, then a
// single-block combiner (no float atomics -> bit-stable across graph replays).
// Cross-lane reduction uses v_wmma_f32_16x16x4_f32 (A = ones) to collapse
// lane n and n+16 partials in one matrix-pipe instruction.
// ---------------------------------------------------------------------------

typedef __attribute__((ext_vector_type(2))) float v2f;
typedef __attribute__((ext_vector_type(8))) float v8f;

// Sum of one float per lane across the wave32, result in every lane.
// WMMA with A=ones(16x4), B=(p,0): D[m,n] = p[n] + p[n+16] for all m,
// so D vgpr0 holds q[lane%16] = p[lane%16] + p[lane%16+16] in all lanes.
// Four xor-butterflies over the 16 columns finish the reduction.
__device__ __forceinline__ float wave_sum32(float p) {
  v2f a; a.x = 1.0f; a.y = 1.0f;       // A-matrix: all ones
  v2f b; b.x = p;    b.y = 0.0f;       // B-matrix: per-lane partial
  v8f c = {};
  c = __builtin_amdgcn_wmma_f32_16x16x4_f32(
      /*neg_a=*/false, a, /*neg_b=*/false, b,
      /*c_mod=*/(short)0, c, /*reuse_a=*/false, /*reuse_b=*/false);
  float q = c[0];
  q += __shfl_xor(q, 1, 32);
  q += __shfl_xor(q, 2, 32);
  q += __shfl_xor(q, 4, 32);
  q += __shfl_xor(q, 8, 32);
  return q;
}

// Block-wide sum (256 threads = 8 waves). Valid on threads of wave 0.
// All WMMA call sites run with full per-wave EXEC.
__device__ __forceinline__ float block_sum(float v, float* sm) {
  const int lane = threadIdx.x & 31;
  const int wid  = threadIdx.x >> 5;
  __syncthreads();                       // protect sm reuse across calls
  if (threadIdx.x < 32) sm[threadIdx.x] = 0.0f;   // wave 0 only
  __syncthreads();
  float w = wave_sum32(v);               // all waves, full EXEC
  if (lane == 0) sm[wid] = w;
  __syncthreads();
  float r = 0.0f;
  if (wid == 0) r = wave_sum32(sm[lane]); // wave-uniform branch, full EXEC
  return r;
}

__device__ __forceinline__ float gate_term(float x, float y) {
  // stable BCEWithLogits: max(x,0) - x*y + log1p(exp(-|x|))
  return fmaxf(x, 0.0f) - x * y + log1pf(__expf(-fabsf(x)));
}

// ---------------------------------------------------------------------------
// Fused partial-sum kernel. Block ranges:
//   [0, G1)            : mel L1 + MSE         -> ws[0..G1), ws[G1..2G1)
//   [G1, G1+G2)        : gate BCE             -> ws[2G1..2G1+G2)
//   [G1+G2, G1+G2+G3)  : guided attention     -> ws[2G1+G2..2G1+G2+G3)
//   G1+G2+G3           : expressive (tiny)    -> ws[2G1+G2+G3]
// ---------------------------------------------------------------------------
extern "C" __global__ void __launch_bounds__(256)
tacoloss_partials(const float* __restrict__ mel_o, const float* __restrict__ mel_p,
                  const float* __restrict__ mel_t, const float* __restrict__ gate_o,
                  const float* __restrict__ gate_t, const float* __restrict__ alignm,
                  const float* __restrict__ e_prob, const int* __restrict__ e_id,
                  float* __restrict__ ws,
                  int n_mel, int n_gate, int T, int N, int B, int C,
                  int G1, int G2, int G3)
{
  __shared__ float sm[32];
  const int bx = blockIdx.x;

  if (bx < G1) {
    // ------------------------- mel L1 + MSE -------------------------------
    const int rb  = bx;
    const int nth = G1 * 256;
    const int tid = rb * 256 + threadIdx.x;
    const int n4  = n_mel >> 2;
    const float4* a4 = (const float4*)mel_o;
    const float4* p4 = (const float4*)mel_p;
    const float4* t4 = (const float4*)mel_t;
    float sl1 = 0.0f, sms = 0.0f;
    for (int i = tid; i < n4; i += nth) {
      float4 av = a4[i], pv = p4[i], tv = t4[i];
      float d;
      d = av.x - tv.x; sl1 += fabsf(d); sms += d * d;
      d = av.y - tv.y; sl1 += fabsf(d); sms += d * d;
      d = av.z - tv.z; sl1 += fabsf(d); sms += d * d;
      d = av.w - tv.w; sl1 += fabsf(d); sms += d * d;
      d = pv.x - tv.x; sl1 += fabsf(d); sms += d * d;
      d = pv.y - tv.y; sl1 += fabsf(d); sms += d * d;
      d = pv.z - tv.z; sl1 += fabsf(d); sms += d * d;
      d = pv.w - tv.w; sl1 += fabsf(d); sms += d * d;
    }
    if (rb == 0) {  // scalar tail (<4 elements when 4-aligned)
      for (int i = (n4 << 2) + threadIdx.x; i < n_mel; i += 256) {
        float t = mel_t[i];
        float d = mel_o[i] - t; sl1 += fabsf(d); sms += d * d;
        d = mel_p[i] - t;       sl1 += fabsf(d); sms += d * d;
      }
    }
    float r1 = block_sum(sl1, sm);
    float r2 = block_sum(sms, sm);
    if (threadIdx.x == 0) { ws[rb] = r1; ws[G1 + rb] = r2; }

  } else if (bx < G1 + G2) {
    // ------------------------- gate BCE -----------------------------------
    const int rb  = bx - G1;
    const int nth = G2 * 256;
    const int tid = rb * 256 + threadIdx.x;
    const int n4  = n_gate >> 2;
    const float4* x4 = (const float4*)gate_o;
    const float4* y4 = (const float4*)gate_t;
    float s = 0.0f;
    for (int i = tid; i < n4; i += nth) {
      float4 x = x4[i], y = y4[i];
      s += gate_term(x.x, y.x) + gate_term(x.y, y.y)
         + gate_term(x.z, y.z) + gate_term(x.w, y.w);
    }
    if (rb == 0) {
      for (int i = (n4 << 2) + threadIdx.x; i < n_gate; i += 256)
        s += gate_term(gate_o[i], gate_t[i]);
    }
    float r = block_sum(s, sm);
    if (threadIdx.x == 0) ws[2 * G1 + rb] = r;

  } else if (bx < G1 + G2 + G3) {
    // --------------------- guided attention -------------------------------
    // sum_{b,t,n} w[t,n] * A[b,t,n]; w computed once per (t,n), reused over b
    const int rb  = bx - (G1 + G2);
    const int nth = G3 * 256;
    const int tid = rb * 256 + threadIdx.x;
    const int TN  = T * N;
    const float invT  = 1.0f / (float)T;
    const float invN  = 1.0f / (float)N;
    const float sharp = 12.5f;  // 1 / (2 * 0.2^2)
    float s = 0.0f;
    if ((TN & 3) == 0) {
      const int tn4 = TN >> 2;
      for (int i = tid; i < tn4; i += nth) {
        const int tn0 = i << 2;
        float wv[4];
        #pragma unroll
        for (int k = 0; k < 4; ++k) {
          int tn = tn0 + k;
          int tt = tn / N;
          int nn = tn - tt * N;
          float dd = (float)nn * invN - (float)tt * invT;
          wv[k] = 1.0f - __expf(-dd * dd * sharp);
        }
        for (int b = 0; b < B; ++b) {
          const float4 v = ((const float4*)(alignm + (size_t)b * TN))[i];
          s += wv[0] * v.x + wv[1] * v.y + wv[2] * v.z + wv[3] * v.w;
        }
      }
    } else {  // generic fallback, scalar
      for (int tn = tid; tn < TN; tn += nth) {
        int tt = tn / N;
        int nn = tn - tt * N;
        float dd = (float)nn * invN - (float)tt * invT;
        float w = 1.0f - __expf(-dd * dd * sharp);
        for (int b = 0; b < B; ++b) s += w * alignm[(size_t)b * TN + tn];
      }
    }
    float r = block_sum(s, sm);
    if (threadIdx.x == 0) ws[2 * G1 + G2 + rb] = r;

  } else {
    // --------------------- expressive (gst) -------------------------------
    // one-hot log-softmax folds: sum_c e_prob[b,c]*log_q[c]
    //   = e_prob[b, id_b] - log(e + C - 1) * rowsum(e_prob[b])
    const float L = logf(__expf(1.0f) + (float)(C - 1));
    float val = 0.0f;
    for (int b = threadIdx.x; b < B; b += 256) {
      float rs = 0.0f;
      for (int cc = 0; cc < C; ++cc) rs += e_prob[b * C + cc];
      val += e_prob[b * C + e_id[b]] - L * rs;
    }
    float r = block_sum(val, sm);
    if (threadIdx.x == 0) ws[2 * G1 + G2 + G3] = r;
  }
}

// ---------------------------------------------------------------------------
// Combiner: one block, double accumulation for bit-stable final scalar.
// loss = (Sl1+Smse)/Nmel + Sgate/Ngate + Salign/Nalign
//        + 0.5 * (Sexpr/B + ln(10))          [cat_lambda == 0.5 always:
//        CAT_STEP==1, increments 0 => lambda schedule is constant]
// ---------------------------------------------------------------------------
extern "C" __global__ void __launch_bounds__(256)
tacoloss_final(const float* __restrict__ ws, float* __restrict__ out,
               int G1, int G2, int G3,
               float inv_nmel, float inv_ngate, float inv_nalign, float inv_B)
{
  __shared__ double sd[256];
  double m = 0.0, g = 0.0, al = 0.0;
  for (int i = threadIdx.x; i < G1; i += 256) m  += (double)ws[i] + (double)ws[G1 + i];
  for (int i = threadIdx.x; i < G2; i += 256) g  += (double)ws[2 * G1 + i];
  for (int i = threadIdx.x; i < G3; i += 256) al += (double)ws[2 * G1 + G2 + i];
  sd[threadIdx.x] = m * (double)inv_nmel + g * (double)inv_ngate + al * (double)inv_nalign;
  __syncthreads();
  for (int st = 128; st > 0; st >>= 1) {
    if ((int)threadIdx.x < st) sd[threadIdx.x] += sd[threadIdx.x + st];
    __syncthreads();
  }
  if (threadIdx.x == 0) {
    double expr = 0.5 * ((double)ws[2 * G1 + G2 + G3] * (double)inv_B
                         + 2.302585092994046 /* -log(0.1) */);
    out[0] = (float)(sd[0] + expr);
  }
}

// ---------------------------------------------------------------------------
extern "C" void kernel_launch(void* const* d_in, const int* in_sizes, int n_in,
                              void* d_out, int out_size, void* d_ws, size_t ws_size,
                              hipStream_t stream) {
  (void)n_in; (void)out_size; (void)ws_size;
  // setup_inputs order:
  // 0:iteration 1:mel_outputs 2:mel_outputs_postnet 3:gate_out 4:alignments
  // 5:e_prob 6:mel_target 7:gate_target 8:e_id
  const float* mel_o  = (const float*)d_in[1];
  const float* mel_p  = (const float*)d_in[2];
  const float* gate_o = (const float*)d_in[3];
  const float* alignm = (const float*)d_in[4];
  const float* e_prob = (const float*)d_in[5];
  const float* mel_t  = (const float*)d_in[6];
  const float* gate_t = (const float*)d_in[7];
  const int*   e_id   = (const int*)d_in[8];

  const int n_mel   = in_sizes[1];            // 32*2000*80 = 5,120,000
  const int n_gate  = in_sizes[3];            // 32*2000    = 64,000
  const int n_align = in_sizes[4];            // 32*2000*400 = 25,600,000
  const int B = in_sizes[8];                  // 32
  const int C = in_sizes[5] / B;              // 8
  const int T = n_gate / B;                   // 2000
  const int N = n_align / (B * T);            // 400

  // block counts ~ proportional to bytes streamed per region
  const int G1 = 1024, G2 = 64, G3 = 800;
  float* ws = (float*)d_ws;                   // (2*G1 + G2 + G3 + 1) floats ~ 11.6 KB

  tacoloss_partials<<<dim3(G1 + G2 + G3 + 1), dim3(256), 0, stream>>>(
      mel_o, mel_p, mel_t, gate_o, gate_t, alignm, e_prob, e_id, ws,
      n_mel, n_gate, T, N, B, C, G1, G2, G3);

  tacoloss_final<<<dim3(1), dim3(256), 0, stream>>>(
      ws, (float*)d_out, G1, G2, G3,
      1.0f / (float)n_mel, 1.0f / (float)n_gate,
      1.0f / (float)n_align, 1.0f / (float)B);
}